// DingoNet_76854144795142
// MI455X (gfx1250) — compile-verified
//
#include <hip/hip_runtime.h>
#include <hip/hip_bf16.h>
#include <math.h>

// ---------------------------------------------------------------------------
// Problem constants (from the reference)
// ---------------------------------------------------------------------------
#define BATCH      128
#define SEQ_LEN    8194
#define NCH        32
#define PL         8192          // SEQ_LEN - 2
#define D          128           // NCH * 4
#define N          2048          // PL / 4
#define NTILES     (N / 16)      // 128
#define EMB_DIM    5
#define KW         3

typedef float v2f __attribute__((ext_vector_type(2)));
typedef float v8f __attribute__((ext_vector_type(8)));

// Async-to-LDS builtin wants pointers to GCC-style int4 vectors:
//   (v4i addrspace(1)*, v4i addrspace(3)*, imm offset, imm cpol)
typedef int async_v4i __attribute__((vector_size(16)));
typedef __attribute__((address_space(1))) async_v4i* g_v4i_ptr;
typedef __attribute__((address_space(3))) async_v4i* l_v4i_ptr;

#ifndef __has_builtin
#define __has_builtin(x) 0
#endif

#if __has_builtin(__builtin_amdgcn_global_load_async_to_lds_b128)
#define USE_ASYNC_LDS 1
#else
#define USE_ASYNC_LDS 0
#endif

__device__ __forceinline__ void wait_asynccnt0() {
#if __has_builtin(__builtin_amdgcn_s_wait_asynccnt)
    __builtin_amdgcn_s_wait_asynccnt(0);
#else
    asm volatile("s_wait_asynccnt 0x0" ::: "memory");
#endif
}

#if USE_ASYNC_LDS
__device__ __forceinline__ void async_copy16(const float* gsrc, float* ldst) {
    __builtin_amdgcn_global_load_async_to_lds_b128(
        (g_v4i_ptr)gsrc, (l_v4i_ptr)ldst, 0, 0);
}
#endif

// ---------------------------------------------------------------------------
// Kernel 1: embedding lookup + conv1d(5->32, k=3, valid) + ReLU
// Writes y[b, c, p] at ws[b*D*N + c*PL + p]  (== Y[b, d, n] after the
// contiguous reshape, since (c*4+s)*2048 + n == c*8192 + p).
// grid = (PL/256, BATCH), block = 256
// ---------------------------------------------------------------------------
__global__ __launch_bounds__(256)
void dingo_embed_conv(const int* __restrict__ seq,
                      const float* __restrict__ emb,
                      const float* __restrict__ conv_w,
                      const float* __restrict__ conv_b,
                      float* __restrict__ Y)
{
    __shared__ float s_emb[26 * EMB_DIM];     // 130
    __shared__ float s_w[NCH * EMB_DIM * KW]; // 480
    __shared__ float s_b[NCH];

    const int t = threadIdx.x;
    const int b = blockIdx.y;
    const int p = blockIdx.x * 256 + t;

    for (int i = t; i < 26 * EMB_DIM; i += 256) s_emb[i] = emb[i];
    for (int i = t; i < NCH * EMB_DIM * KW; i += 256) s_w[i] = conv_w[i];
    if (t < NCH) s_b[t] = conv_b[t];
    __syncthreads();

    const int base = b * SEQ_LEN + p;
    const int s0 = seq[base + 0];
    const int s1 = seq[base + 1];
    const int s2 = seq[base + 2];
    const float* e0 = s_emb + s0 * EMB_DIM;
    const float* e1 = s_emb + s1 * EMB_DIM;
    const float* e2 = s_emb + s2 * EMB_DIM;

    float* yb = Y + (size_t)b * (D * N) + p;
#pragma unroll 4
    for (int c = 0; c < NCH; ++c) {
        const float* wc = s_w + c * (EMB_DIM * KW);
        float acc = s_b[c];
#pragma unroll
        for (int e = 0; e < EMB_DIM; ++e) {
            acc += wc[e * 3 + 0] * e0[e];
            acc += wc[e * 3 + 1] * e1[e];
            acc += wc[e * 3 + 2] * e2[e];
        }
        yb[c * PL] = fmaxf(acc, 0.0f);
    }
}

// ---------------------------------------------------------------------------
// Kernel 2: per-batch attention pooling.
//   T = tanh(W_a @ Y_b)   via V_WMMA_F32_16X16X4_F32
//   e = v_a . T ; w = softmax(e) ; ctx = Y_b @ w
// grid = BATCH, block = 256 (8 waves; wave w owns rows 16w..16w+15)
// Double-buffered LDS tile; next tile's fill overlaps the WMMA loop via
// global_load_async_to_lds_b128 (ASYNCcnt) when available.
// ---------------------------------------------------------------------------
__global__ __launch_bounds__(256)
void dingo_attn(const float* __restrict__ Y,
                const float* __restrict__ W_a,
                const float* __restrict__ v_a,
                float* __restrict__ out)
{
    __shared__ float lds_y[2][D * 16]; // double-buffered tile, 2 x 8KB
    __shared__ float e_lds[N];         // energies -> exp weights, 8KB
    __shared__ float red[256];

    const int t    = threadIdx.x;
    const int b    = blockIdx.x;
    const int lane = t & 31;
    const int w    = t >> 5;          // wave id, 0..7
    const int col  = lane & 15;
    const int hi   = lane >> 4;       // 0: lanes 0-15, 1: lanes 16-31

    const float* Yb = Y + (size_t)b * (D * N);

    // staging assignment: 2 threads per row, 8 floats (2 x b128) each
    const int st_d  = t >> 1;
    const int st_h  = (t & 1) * 8;
    const float* st_src = Yb + (size_t)st_d * N + st_h;
    float* st_dst0 = &lds_y[0][st_d * 16 + st_h];
    float* st_dst1 = &lds_y[1][st_d * 16 + st_h];

    // --- preload A fragments: W_a rows [16w .. 16w+15], all K, into regs ----
    // lane l<16 holds W_a[16w+l, 4k+0..1]; lanes 16-31 hold cols 4k+2..3
    const float* wa_row = W_a + (w * 16 + col) * D + 2 * hi;
    float2 a_frag[32];
#pragma unroll
    for (int k = 0; k < 32; ++k)
        a_frag[k] = *(const float2*)(wa_row + 4 * k);

    // v_a slice for this lane's 8 rows
    float va[8];
#pragma unroll
    for (int r = 0; r < 8; ++r)
        va[r] = v_a[w * 16 + hi * 8 + r];

    // zero the energy accumulator
    for (int n = t; n < N; n += 256) e_lds[n] = 0.0f;

    // --- prologue: stage tile 0 into buffer 0 ---------------------------
#if USE_ASYNC_LDS
    async_copy16(st_src,     st_dst0);
    async_copy16(st_src + 4, st_dst0 + 4);
    wait_asynccnt0();
#else
    {
        float4 v0 = ((const float4*)st_src)[0];
        float4 v1 = ((const float4*)st_src)[1];
        *(float4*)(st_dst0 + 0) = v0;
        *(float4*)(st_dst0 + 4) = v1;
    }
#endif
    __syncthreads();  // tile 0 staged + e_lds zeroed

    // --- main loop over 128 column tiles of 16 ---------------------------
    for (int tile = 0; tile < NTILES; ++tile) {
        const int  cur       = tile & 1;
        const bool have_next = (tile + 1) < NTILES;
        const float* nsrc    = st_src + (size_t)(tile + 1) * 16;
        float* ndst          = cur ? st_dst0 : st_dst1;
        const float* buf     = lds_y[cur];

        // kick off next tile's fill so it overlaps the WMMA loop
#if USE_ASYNC_LDS
        if (have_next) {
            async_copy16(nsrc,     ndst);
            async_copy16(nsrc + 4, ndst + 4);
        }
#else
        float4 nv0, nv1;
        if (have_next) {
            nv0 = ((const float4*)nsrc)[0];
            nv1 = ((const float4*)nsrc)[1];
        }
#endif

        // --- GEMM: T = W_a[16w:16w+16, :] @ Ytile (two acc chains) ------
        v8f acc0 = {};
        v8f acc1 = {};
#pragma unroll
        for (int k = 0; k < 32; k += 2) {
            {
                const int r0 = 4 * k + 2 * hi;
                v2f a;  a.x = a_frag[k].x;      a.y = a_frag[k].y;
                v2f bb; bb.x = buf[r0 * 16 + col];
                        bb.y = buf[r0 * 16 + 16 + col];
                acc0 = __builtin_amdgcn_wmma_f32_16x16x4_f32(
                           false, a, false, bb, (short)0, acc0, false, false);
            }
            {
                const int r0 = 4 * (k + 1) + 2 * hi;
                v2f a;  a.x = a_frag[k + 1].x;  a.y = a_frag[k + 1].y;
                v2f bb; bb.x = buf[r0 * 16 + col];
                        bb.y = buf[r0 * 16 + 16 + col];
                acc1 = __builtin_amdgcn_wmma_f32_16x16x4_f32(
                           false, a, false, bb, (short)0, acc1, false, false);
            }
        }
        v8f acc = acc0 + acc1;

        // --- energies: partial = sum_r v_a[row] * tanh(T[row, col]) -----
        float partial = 0.0f;
#pragma unroll
        for (int r = 0; r < 8; ++r)
            partial += va[r] * tanhf(acc[r]);
        partial += __shfl_xor(partial, 16, 32);   // fold upper 8 rows
        if (lane < 16)
            atomicAdd(&e_lds[tile * 16 + lane], partial); // ds_add_f32

#if USE_ASYNC_LDS
        wait_asynccnt0();
#else
        if (have_next) {
            *(float4*)(ndst + 0) = nv0;
            *(float4*)(ndst + 4) = nv1;
        }
#endif
        __syncthreads();  // next tile staged, current tile's readers done
    }

    // --- softmax over e_lds[0..N) (stable) ------------------------------
    float mx = -INFINITY;
    for (int n = t; n < N; n += 256) mx = fmaxf(mx, e_lds[n]);
    red[t] = mx;
    __syncthreads();
    for (int s = 128; s > 0; s >>= 1) {
        if (t < s) red[t] = fmaxf(red[t], red[t + s]);
        __syncthreads();
    }
    mx = red[0];
    __syncthreads();

    float sum = 0.0f;
    for (int n = t; n < N; n += 256) {
        float ex = expf(e_lds[n] - mx);
        e_lds[n] = ex;
        sum += ex;
    }
    red[t] = sum;
    __syncthreads();
    for (int s = 128; s > 0; s >>= 1) {
        if (t < s) red[t] += red[t + s];
        __syncthreads();
    }
    const float inv = 1.0f / red[0];
    __syncthreads();

    // --- ctx[b, d] = inv * sum_n e_lds[n] * Y[b, d, n] -------------------
    // wave w handles rows 16w..16w+15; coalesced 32-lane sweeps over n
#pragma unroll 1
    for (int i = 0; i < 16; ++i) {
        const int r = w * 16 + i;
        const float* yr = Yb + (size_t)r * N;
        float acc = 0.0f;
        for (int n = lane; n < N; n += 32)
            acc += e_lds[n] * yr[n];
#pragma unroll
        for (int off = 16; off > 0; off >>= 1)
            acc += __shfl_xor(acc, off, 32);
        if (lane == 0)
            out[b * D + r] = acc * inv;
    }
}

// ---------------------------------------------------------------------------
extern "C" void kernel_launch(void* const* d_in, const int* in_sizes, int n_in,
                              void* d_out, int out_size, void* d_ws, size_t ws_size,
                              hipStream_t stream)
{
    const int*   seq    = (const int*)  d_in[0];
    const float* emb    = (const float*)d_in[1];
    const float* conv_w = (const float*)d_in[2];
    const float* conv_b = (const float*)d_in[3];
    const float* W_a    = (const float*)d_in[4];
    const float* v_a    = (const float*)d_in[5];
    float* out = (float*)d_out;
    float* Y   = (float*)d_ws;   // 128*128*2048*4 = 128 MB

    dingo_embed_conv<<<dim3(PL / 256, BATCH), 256, 0, stream>>>(
        seq, emb, conv_w, conv_b, Y);
    dingo_attn<<<BATCH, 256, 0, stream>>>(Y, W_a, v_a, out);
}